// LTQ_82798379532993
// MI455X (gfx1250) — compile-verified
//
#include <hip/hip_runtime.h>

#define LTQ_EPS  1e-3f
#define LTQ_NV   15
#define LTQ_INT  (2.0f / 15.0f)

typedef float v4f __attribute__((ext_vector_type(4)));
typedef int   v4i_vs __attribute__((vector_size(16)));   // matches builtin param type

#define LTQ_AS1 __attribute__((address_space(1)))
#define LTQ_AS3 __attribute__((address_space(3)))

// CDNA5 async global->LDS path (ASYNCcnt), guarded so absence can't break compile.
#if defined(__AMDGCN__) && __has_builtin(__builtin_amdgcn_global_load_async_to_lds_b128)
#define LTQ_ASYNC_LDS 1
#else
#define LTQ_ASYNC_LDS 0
#endif

// Workspace layout (floats):
//  [0..14]   tf[i]            (forward thresholds, uniform -> SGPRs)
//  [15]      top              (s + cums[14])
//  [16]      s1 (scale1)
//  [17]      s2 (scale2)
//  [32..95]  T1: 16 x float4  {tb[k], A[k], B[k], A[k+1]}   (tb[15] = +inf)
//  [96..111] T2: 16 x float   {B[k+1]}
//
// Piecewise-affine backward branch: for j = cnt_b in [0,15],
//   xb = A[j]*xs + B[j],  A[0]=B[0]=0 (cnt_b==0 -> 0),
//   A[j] = INTERVAL/a_pos[j-1],  B[j] = (j-1)*INTERVAL - A[j]*tb[j-1].
// Interleaving tb[k] < tf[k] < tb[k+1] gives cnt_b = cnt_f + (xs > tb[cnt_f]).

__global__ void ltq_setup_kernel(const float* __restrict__ start,
                                 const float* __restrict__ a,
                                 const float* __restrict__ scale1,
                                 const float* __restrict__ scale2,
                                 float* __restrict__ ws) {
  if (threadIdx.x != 0 || blockIdx.x != 0) return;
  const float s = start[0];

  float ap[LTQ_NV], cums[LTQ_NV], tb[LTQ_NV + 1], tf[LTQ_NV];
  for (int i = 0; i < LTQ_NV; ++i) {
    float ai = a[i];
    ap[i] = (ai > LTQ_EPS) ? ai : LTQ_EPS;
  }
  // sequential cumsum, same association as jnp.cumsum
  float c = 0.0f;
  for (int i = 0; i < LTQ_NV; ++i) { c += ap[i]; cums[i] = c; }
  tb[0] = s;
  for (int i = 1; i < LTQ_NV; ++i) tb[i] = s + cums[i - 1];
  tb[LTQ_NV] = __builtin_inff();                 // sentinel: k==15 -> never "up"
  for (int i = 0; i < LTQ_NV; ++i) tf[i] = tb[i] + ap[i] * 0.5f;
  const float top = s + cums[LTQ_NV - 1];

  float A[LTQ_NV + 1], B[LTQ_NV + 1];
  A[0] = 0.0f; B[0] = 0.0f;
  for (int j = 1; j <= LTQ_NV; ++j) {
    int idx = j - 1;
    float inv = LTQ_INT / ap[idx];
    A[j] = inv;
    B[j] = (float)idx * LTQ_INT - inv * tb[idx];
  }

  for (int i = 0; i < LTQ_NV; ++i) ws[i] = tf[i];
  ws[15] = top;
  ws[16] = scale1[0];
  ws[17] = scale2[0];
  for (int k = 0; k <= LTQ_NV; ++k) {
    float A1 = (k < LTQ_NV) ? A[k + 1] : A[LTQ_NV];
    float B1 = (k < LTQ_NV) ? B[k + 1] : B[LTQ_NV];
    ws[32 + 4 * k + 0] = tb[k];
    ws[32 + 4 * k + 1] = A[k];
    ws[32 + 4 * k + 2] = B[k];
    ws[32 + 4 * k + 3] = A1;
    ws[96 + k]         = B1;
  }
}

__device__ __forceinline__ float ltq_elem(float x, float s1, float s2, float top,
                                          const float* __restrict__ tfv,
                                          const v4f* T1s, const float* T2s) {
  float xs = x * s1;
  int k = 0;
#pragma unroll
  for (int i = 0; i < LTQ_NV; ++i) k += (xs > tfv[i]) ? 1 : 0;   // cnt_f in [0,15]
  v4f   e  = T1s[k];             // {tb[k], A[k], B[k], A[k+1]} — conflict-free banks
  float b2 = T2s[k];             // B[k+1]
  bool up = xs > e.x;            // tb[k] < xs  ->  cnt_b = k+1
  float A = up ? e.w : e.y;
  float B = up ? b2  : e.z;
  float xb = fmaf(A, xs, B);
  xb = (xs > top) ? 2.0f : xb;
  float xf = (float)k * LTQ_INT;
  return ((xf + xb) - xb) * s2;  // literal straight-through combine
}

__global__ __launch_bounds__(256) void ltq_main_kernel(
    const float* __restrict__ x, float* __restrict__ out,
    const float* __restrict__ ws, long long n) {
  __shared__ __align__(16) v4f   T1s[16];
  __shared__ __align__(16) float T2s[16];
  const int t = threadIdx.x;

#if LTQ_ASYNC_LDS
  // CDNA5: DMA the 320-byte LUT straight into LDS (no VGPR round-trip),
  // tracked by ASYNCcnt. 16 lanes x b128 for T1, 4 lanes x b128 for T2.
  if (t < 16) {
    __builtin_amdgcn_global_load_async_to_lds_b128(
        (LTQ_AS1 v4i_vs*)(ws + 32 + 4 * t),
        (LTQ_AS3 v4i_vs*)&T1s[t], 0, 0);
  }
  if (t < 4) {
    __builtin_amdgcn_global_load_async_to_lds_b128(
        (LTQ_AS1 v4i_vs*)(ws + 96 + 4 * t),
        (LTQ_AS3 v4i_vs*)&T2s[4 * t], 0, 0);
  }
#else
  if (t < 16) {
    T1s[t] = ((const v4f*)(ws + 32))[t];
    T2s[t] = ws[96 + t];
  }
#endif

  float tfv[LTQ_NV];
#pragma unroll
  for (int i = 0; i < LTQ_NV; ++i) tfv[i] = ws[i];   // uniform -> SGPRs
  const float top = ws[15];
  const float s1  = ws[16];
  const float s2  = ws[17];

#if LTQ_ASYNC_LDS
#if __has_builtin(__builtin_amdgcn_s_wait_asynccnt)
  __builtin_amdgcn_s_wait_asynccnt(0);
#else
  asm volatile("s_wait_asynccnt 0" ::: "memory");
#endif
#endif
  __syncthreads();

  const long long n4 = n >> 2;
  const v4f* __restrict__ x4 = (const v4f*)x;
  v4f* __restrict__ o4 = (v4f*)out;
  const long long stride = (long long)blockDim.x * gridDim.x;

  for (long long i = (long long)blockIdx.x * blockDim.x + t; i < n4; i += stride) {
    long long ip = i + stride;                       // gfx1250 global_prefetch_b8
    if (ip < n4) __builtin_prefetch((const void*)&x4[ip], 0, 1);
    v4f v = x4[i];
    v4f r;
    r.x = ltq_elem(v.x, s1, s2, top, tfv, T1s, T2s);
    r.y = ltq_elem(v.y, s1, s2, top, tfv, T1s, T2s);
    r.z = ltq_elem(v.z, s1, s2, top, tfv, T1s, T2s);
    r.w = ltq_elem(v.w, s1, s2, top, tfv, T1s, T2s);
    __builtin_nontemporal_store(r, &o4[i]);          // write-once stream: NT hint
  }

  // tail (n not multiple of 4 — not hit for 25,690,112 but kept for safety)
  const long long tail = n & 3LL;
  if (tail) {
    long long gid = (long long)blockIdx.x * blockDim.x + t;
    long long base = n4 << 2;
    if (gid < tail)
      out[base + gid] = ltq_elem(x[base + gid], s1, s2, top, tfv, T1s, T2s);
  }
}

extern "C" void kernel_launch(void* const* d_in, const int* in_sizes, int n_in,
                              void* d_out, int out_size, void* d_ws, size_t ws_size,
                              hipStream_t stream) {
  const float* x      = (const float*)d_in[0];
  const float* start  = (const float*)d_in[1];
  const float* a      = (const float*)d_in[2];
  const float* scale1 = (const float*)d_in[3];
  const float* scale2 = (const float*)d_in[4];
  float* out = (float*)d_out;
  float* ws  = (float*)d_ws;

  const long long n = (long long)in_sizes[0];

  ltq_setup_kernel<<<1, 32, 0, stream>>>(start, a, scale1, scale2, ws);

  const int threads = 256;                 // 8 wave32 waves per block
  long long n4 = (n >> 2);
  if (n4 == 0) n4 = 1;
  long long want = (n4 + threads - 1) / threads;
  long long cap  = 6272;                   // 6272*256*4*4 == 25,690,112: 4 iters/thread
  int blocks = (int)((want < cap) ? want : cap);
  if (blocks < 1) blocks = 1;

  ltq_main_kernel<<<blocks, threads, 0, stream>>>(x, out, ws, n);
}